// SinusoidalEmbedding_4389456576519
// MI455X (gfx1250) — compile-verified
//
#include <hip/hip_runtime.h>

// Sinusoidal embedding: out[row, 2i] = sin(x[row] * f_i), out[row, 2i+1] = cos(...)
// f_i = 10000^(-2i/256), DIM = 256, HALF = 128.
//
// Layout: one wave32 per row. Lane L handles i = 4L..4L+3, producing 8
// consecutive floats at out + row*256 + 8L  -> two 16-byte NT stores per lane,
// one perfectly contiguous 1 KB row per wave. Store-bandwidth bound (~1 GiB
// output); NT hint keeps the write-once stream from thrashing L2 (192 MB < 1 GiB).

typedef __attribute__((ext_vector_type(4))) float v4f;

#define NEG_LOG2_BASE_OVER_HALF (-0.103810253f)   // -log2(10000)/128
#define FREQ_RATIO              (0.93057204f)     // 2^(-log2(10000)/128)
#define INV_2PI                 (0.15915494309189535f)

__device__ __forceinline__ float fast_exp2(float v) {
#if __has_builtin(__builtin_amdgcn_exp2f)
  return __builtin_amdgcn_exp2f(v);   // v_exp_f32
#else
  return exp2f(v);                    // HIP device libm fallback
#endif
}

__global__ __launch_bounds__(256) void sinembed_kernel(
    const float* __restrict__ x, float* __restrict__ out, unsigned n) {
  const unsigned g    = blockIdx.x * 256u + threadIdx.x;
  const unsigned row  = g >> 5;     // one wave (32 lanes) per row
  const unsigned lane = g & 31u;
  if (row >= n) return;

  const float xv = x[row];          // uniform across the wave (broadcast load)

  // inv_freq with 1/(2pi) folded in, so V_SIN/V_COS get pre-normalized angles.
  const unsigned i0 = lane * 4u;
  const float f0 = fast_exp2((float)i0 * NEG_LOG2_BASE_OVER_HALF) * INV_2PI;
  const float f1 = f0 * FREQ_RATIO;
  const float f2 = f1 * FREQ_RATIO;
  const float f3 = f2 * FREQ_RATIO;

  const float e0 = xv * f0;
  const float e1 = xv * f1;
  const float e2 = xv * f2;
  const float e3 = xv * f3;

  v4f o0, o1;
  o0.x = __builtin_amdgcn_sinf(e0);
  o0.y = __builtin_amdgcn_cosf(e0);
  o0.z = __builtin_amdgcn_sinf(e1);
  o0.w = __builtin_amdgcn_cosf(e1);
  o1.x = __builtin_amdgcn_sinf(e2);
  o1.y = __builtin_amdgcn_cosf(e2);
  o1.z = __builtin_amdgcn_sinf(e3);
  o1.w = __builtin_amdgcn_cosf(e3);

  // float offset row*256 + 8*lane  ==  float4 offset row*64 + 2*lane
  v4f* outv = (v4f*)out + ((size_t)row * 64u + lane * 2u);
  __builtin_nontemporal_store(o0, outv);
  __builtin_nontemporal_store(o1, outv + 1);
}

extern "C" void kernel_launch(void* const* d_in, const int* in_sizes, int n_in,
                              void* d_out, int out_size, void* d_ws, size_t ws_size,
                              hipStream_t stream) {
  (void)n_in; (void)out_size; (void)d_ws; (void)ws_size;
  const float* x = (const float*)d_in[0];
  float* out     = (float*)d_out;
  const unsigned n = (unsigned)in_sizes[0];        // 1,048,576 rows
  const unsigned total_threads = n * 32u;          // one wave32 per row
  const unsigned blocks = (total_threads + 255u) / 256u;
  sinembed_kernel<<<blocks, 256, 0, stream>>>(x, out, n);
}